// GuidedFilter_70660801954506
// MI455X (gfx1250) — compile-verified
//
#include <hip/hip_runtime.h>

// ---------------------------------------------------------------------------
// Fused guided filter for MI455X (gfx1250, wave32).
// Box blurs are executed as banded GEMMs on V_WMMA_F32_16X16X4_F32 tiles.
// One block = one 32x32 output tile x 16 channels; raw halo (r=10) in LDS.
// ---------------------------------------------------------------------------

#define IMG      512
#define CH       16
#define TS       32          // output tile
#define RAWN     52          // TS + 4*RADIUS
#define RAWROWS  64          // padded to 4 M-blocks of 16
#define RAWPITCH 66          // even, 33 bank-pairs -> conflict-free strided reads
#define ABROWS   48          // a/b region (42 valid) padded to 3 blocks
#define ABPITCH  50
#define TROWS    64          // H-pass output scratch
#define TPITCH   50
#define MAPITCH  34

typedef __attribute__((ext_vector_type(2))) float v2f;
typedef __attribute__((ext_vector_type(8))) float v8f;

__device__ __forceinline__ int refl(int p) {       // jnp.pad mode='reflect'
  p = p < 0 ? -p : p;
  return p >= IMG ? 2 * IMG - 2 - p : p;
}

// Horizontal banded blur: dst[m][n] = sum_k A(src)[m][16*nj+k] * band[k][n]
// MODE 0: A=s0   MODE 1: A=s0*s0   MODE 2: A=s0*s1
template <int MODE>
__device__ __forceinline__ void hblur(const float* __restrict__ s0,
                                      const float* __restrict__ s1, int sPitch,
                                      float* __restrict__ dst, int dPitch,
                                      int mBlocks, int nBlocks,
                                      const v2f* __restrict__ band,
                                      int lane, int wave) {
  const int m = lane & 15, half = lane >> 4;
  const int nT = mBlocks * nBlocks;
  for (int t = wave; t < nT; t += 8) {
    const int mi = t / nBlocks, nj = t - mi * nBlocks;
    const float* p0 = s0 + (mi * 16 + m) * sPitch + nj * 16 + 2 * half;
    const float* p1 = s1 + (mi * 16 + m) * sPitch + nj * 16 + 2 * half;
    v8f c = {};
#pragma unroll
    for (int kt = 0; kt < 7; ++kt) {
      float x0 = p0[4 * kt], x1 = p0[4 * kt + 1];
      if (MODE == 1) { x0 *= x0; x1 *= x1; }
      if (MODE == 2) { x0 *= p1[4 * kt]; x1 *= p1[4 * kt + 1]; }
      v2f a; a.x = x0; a.y = x1;
      c = __builtin_amdgcn_wmma_f32_16x16x4_f32(false, a, false, band[kt],
                                                (short)0, c, false, false);
    }
    float* pd = dst + (mi * 16 + half * 8) * dPitch + nj * 16 + m;
#pragma unroll
    for (int v = 0; v < 8; ++v) pd[v * dPitch] = c[v];
  }
}

// Vertical banded blur: dst[m][n] = sum_k band[m][k] * src[16*mi+k][16*nj+n]
__device__ __forceinline__ void vblur(const float* __restrict__ src, int sPitch,
                                      float* __restrict__ dst, int dPitch,
                                      int mBlocks, int nBlocks,
                                      const v2f* __restrict__ band,
                                      int lane, int wave) {
  const int n = lane & 15, half = lane >> 4;
  const int nT = mBlocks * nBlocks;
  for (int t = wave; t < nT; t += 8) {
    const int mi = t / nBlocks, nj = t - mi * nBlocks;
    const float* ps = src + (mi * 16 + 2 * half) * sPitch + nj * 16 + n;
    v8f c = {};
#pragma unroll
    for (int kt = 0; kt < 7; ++kt) {
      v2f b;
      b.x = ps[(4 * kt) * sPitch];
      b.y = ps[(4 * kt + 1) * sPitch];
      c = __builtin_amdgcn_wmma_f32_16x16x4_f32(false, band[kt], false, b,
                                                (short)0, c, false, false);
    }
    float* pd = dst + (mi * 16 + half * 8) * dPitch + nj * 16 + n;
#pragma unroll
    for (int v = 0; v < 8; ++v) pd[v * dPitch] = c[v];
  }
}

__global__ __launch_bounds__(256) void guided_filter_kernel(
    const float* __restrict__ feat, const float* __restrict__ img,
    float* __restrict__ out) {
  __shared__ float rawI[RAWROWS * RAWPITCH];
  __shared__ float rawF[RAWROWS * RAWPITCH];
  __shared__ float tmpH[TROWS * TPITCH];
  __shared__ float mI [ABROWS * ABPITCH];
  __shared__ float vIi[ABROWS * ABPITCH];
  __shared__ float mP [ABROWS * ABPITCH];
  __shared__ float cIp[ABROWS * ABPITCH];
  __shared__ float mA [TS * MAPITCH];
  __shared__ float mB [TS * MAPITCH];

  const int tid  = threadIdx.x;
  const int lane = tid & 31;
  const int wave = tid >> 5;
  const int x0 = blockIdx.x * TS, y0 = blockIdx.y * TS;
  const int bat = blockIdx.z;

  // Band operand (1/11 inside the 11-wide band), per the f32 WMMA VGPR layout:
  // element index q = lane&15 (N for H-pass B, M for V-pass A),
  // K = 4*kt + vgpr + 2*(lane>>4).  Same registers serve both passes.
  v2f band[7];
  {
    const int q = lane & 15, h2 = (lane >> 4) * 2;
#pragma unroll
    for (int kt = 0; kt < 7; ++kt) {
      const int k0 = 4 * kt + h2;
      band[kt].x = (k0 >= q && k0 <= q + 10) ? (1.0f / 11.0f) : 0.0f;
      band[kt].y = (k0 + 1 >= q && k0 + 1 <= q + 10) ? (1.0f / 11.0f) : 0.0f;
    }
  }

  // Zero raw buffers once (pad rows/cols must be 0 so padded M/K blocks vanish)
  for (int i = tid; i < RAWROWS * RAWPITCH; i += 256) { rawI[i] = 0.0f; rawF[i] = 0.0f; }
  __syncthreads();

  // Load guidance tile (52x52 halo) with reflect indexing
  const float* imgB = img + (size_t)bat * IMG * IMG;
  for (int i = tid; i < RAWN * RAWN; i += 256) {
    const int r = i / RAWN, cc = i - r * RAWN;
    rawI[r * RAWPITCH + cc] = imgB[refl(y0 - 10 + r) * IMG + refl(x0 - 10 + cc)];
  }
  __syncthreads();

  // mean_I
  hblur<0>(rawI, rawI, RAWPITCH, tmpH, TPITCH, 4, 3, band, lane, wave);
  __syncthreads();
  vblur(tmpH, TPITCH, mI, ABPITCH, 3, 3, band, lane, wave);
  __syncthreads();
  // corr_I
  hblur<1>(rawI, rawI, RAWPITCH, tmpH, TPITCH, 4, 3, band, lane, wave);
  __syncthreads();
  vblur(tmpH, TPITCH, vIi, ABPITCH, 3, 3, band, lane, wave);
  __syncthreads();
  // vIi <- 1 / (var_I + eps)
  for (int i = tid; i < ABROWS * ABROWS; i += 256) {
    const int r = i / ABROWS, cc = i - r * ABROWS;
    const int o = r * ABPITCH + cc;
    const float m = mI[o];
    vIi[o] = 1.0f / (vIi[o] - m * m + 1e-8f);
  }
  __syncthreads();

  for (int ch = 0; ch < CH; ++ch) {
    const float* featB = feat + ((size_t)bat * CH + ch) * IMG * IMG;
    for (int i = tid; i < RAWN * RAWN; i += 256) {
      const int r = i / RAWN, cc = i - r * RAWN;
      rawF[r * RAWPITCH + cc] = featB[refl(y0 - 10 + r) * IMG + refl(x0 - 10 + cc)];
    }
    __syncthreads();

    // mean_p
    hblur<0>(rawF, rawF, RAWPITCH, tmpH, TPITCH, 4, 3, band, lane, wave);
    __syncthreads();
    vblur(tmpH, TPITCH, mP, ABPITCH, 3, 3, band, lane, wave);
    __syncthreads();
    // corr_Ip  (product folded into A-operand fetch)
    hblur<2>(rawF, rawI, RAWPITCH, tmpH, TPITCH, 4, 3, band, lane, wave);
    __syncthreads();
    vblur(tmpH, TPITCH, cIp, ABPITCH, 3, 3, band, lane, wave);
    __syncthreads();

    // a = cov/(var+eps) -> cIp ; b = mean_p - a*mean_I -> mP
    for (int i = tid; i < ABROWS * ABROWS; i += 256) {
      const int r = i / ABROWS, cc = i - r * ABROWS;
      const int o = r * ABPITCH + cc;
      const float mi_ = mI[o], mp_ = mP[o];
      const float a = (cIp[o] - mi_ * mp_) * vIi[o];
      cIp[o] = a;
      mP[o]  = mp_ - a * mi_;
    }
    __syncthreads();

    // mean_a (42x42 valid -> 32x32)
    hblur<0>(cIp, cIp, ABPITCH, tmpH, TPITCH, 3, 2, band, lane, wave);
    __syncthreads();
    vblur(tmpH, TPITCH, mA, MAPITCH, 2, 2, band, lane, wave);
    __syncthreads();
    // mean_b
    hblur<0>(mP, mP, ABPITCH, tmpH, TPITCH, 3, 2, band, lane, wave);
    __syncthreads();
    vblur(tmpH, TPITCH, mB, MAPITCH, 2, 2, band, lane, wave);
    __syncthreads();

    // out = mean_a * I + mean_b
    float* outB = out + ((size_t)bat * CH + ch) * IMG * IMG;
    for (int i = tid; i < TS * TS; i += 256) {
      const int y = i >> 5, x = i & 31;
      outB[(size_t)(y0 + y) * IMG + (x0 + x)] =
          mA[y * MAPITCH + x] * rawI[(y + 10) * RAWPITCH + (x + 10)] +
          mB[y * MAPITCH + x];
    }
    __syncthreads();
  }
}

extern "C" void kernel_launch(void* const* d_in, const int* in_sizes, int n_in,
                              void* d_out, int out_size, void* d_ws, size_t ws_size,
                              hipStream_t stream) {
  const float* feat = (const float*)d_in[0];
  const float* img  = (const float*)d_in[1];
  float* out = (float*)d_out;
  const int B = in_sizes[1] / (IMG * IMG);   // guidance is [B,1,H,W]
  dim3 grid(IMG / TS, IMG / TS, B);
  guided_filter_kernel<<<grid, 256, 0, stream>>>(feat, img, out);
}